// AttnCutLoss_25632364822591
// MI455X (gfx1250) — compile-verified
//
#include <hip/hip_runtime.h>
#include <hip/hip_bf16.h>
#include <stdint.h>

#define TAU_F 0.95f

static constexpr int kB       = 8192;
static constexpr int kL       = 2048;
static constexpr int kThreads = 256;
static constexpr int kPT      = kL / kThreads;   // 8 elements per thread
static constexpr int kWaves   = kThreads / 32;   // 8 wave32 per block

typedef unsigned int u32x4 __attribute__((ext_vector_type(4)));
typedef int          i32x4 __attribute__((ext_vector_type(4)));
typedef int          i32x8 __attribute__((ext_vector_type(8)));
typedef float        f32x4 __attribute__((ext_vector_type(4)));

#if __has_builtin(__builtin_amdgcn_tensor_load_to_lds)
#define ATTNCUT_USE_TDM 1
#else
#define ATTNCUT_USE_TDM 0
#endif

// Kernel 1: one block per row. TDM stages the labels slice per wave into LDS,
// overlapped with direct global streaming of the output row + v_log_f32.
__global__ __launch_bounds__(kThreads) void attncut_row_kernel(
    const float* __restrict__ output, const float* __restrict__ labels,
    float* __restrict__ rowpart)
{
  __shared__ float s_lab[kL];          // 8 KB staged labels row (TDM destination)
  __shared__ float s_wsum[kWaves];     // per-wave scan totals
  __shared__ float s_red[2 * kWaves];  // block reduction: sumExp | dot

  const int row  = blockIdx.x;
  const int tid  = threadIdx.x;
  const int lane = tid & 31;
  const int wave = tid >> 5;

  const float* gl = labels + (size_t)row * kL;
  const float* go = output + (size_t)row * kL;

#if ATTNCUT_USE_TDM
  {
    // Each wave DMAs its 256-float (1 KB) slice of the labels row to LDS.
    const int slice = kL / kWaves;  // 256
    uint64_t ga = (uint64_t)(uintptr_t)(gl + wave * slice);
    // Generic pointers to LDS carry the LDS byte offset in addr[31:0] (ISA 10.2).
    uint32_t laddr = (uint32_t)(uintptr_t)(void*)s_lab + (uint32_t)(wave * slice * 4);

    u32x4 g0;
    g0[0] = 1u;                                   // count=1 valid descriptor
    g0[1] = laddr;                                // lds_addr (bytes)
    g0[2] = (uint32_t)ga;                         // global_addr[31:0]
    g0[3] = (uint32_t)((ga >> 32) & 0x1FFFFFFu)   // global_addr[56:32]
          | (2u << 30);                           // type = 2 ("image")
    i32x8 g1;
    g1[0] = (int)(2u << 16);                      // mask=0, data_size=2 (4B)
    g1[1] = (int)((uint32_t)slice << 16);         // tensor_dim0[15:0]=256
    g1[2] = (int)(1u << 16);                      // tensor_dim0 hi=0 | tensor_dim1=1
    g1[3] = (int)((uint32_t)slice << 16);         // tensor_dim1 hi=0 | tile_dim0=256
    g1[4] = 1;                                    // tile_dim1=1, tile_dim2=0
    g1[5] = slice;                                // tensor_dim0_stride lo32
    g1[6] = (int)((uint32_t)slice << 16);         // stride0 hi=0 | stride1 lo=256
    g1[7] = 0;
    i32x4 gz4; gz4[0] = gz4[1] = gz4[2] = gz4[3] = 0;   // groups 2/3 unused (<=2D)
    i32x8 gz8;
    gz8[0] = gz8[1] = gz8[2] = gz8[3] = 0;
    gz8[4] = gz8[5] = gz8[6] = gz8[7] = 0;
    __builtin_amdgcn_tensor_load_to_lds(g0, g1, gz4, gz4, gz8, 0);
  }
#endif

  // Stream the output row directly. These HBM loads must be in flight
  // concurrently with the labels DMA.
  const f32x4* go4 = (const f32x4*)go;
  f32x4 o0 = go4[tid * 2 + 0];
  f32x4 o1 = go4[tid * 2 + 1];

#if ATTNCUT_USE_TDM
  // Full 128-bit vectors as data operands: both global_load_b128 results must
  // materialize before the tensorcnt wait, so the DMA and the wave's own HBM
  // loads overlap; the LDS reads below cannot float above the fence.
  asm volatile("s_wait_tensorcnt 0x0"
               :
               : "v"(o0), "v"(o1)
               : "memory");
  const f32x4* l4 = (const f32x4*)s_lab;
#else
  const f32x4* l4 = (const f32x4*)gl;
#endif
  f32x4 b0 = l4[tid * 2 + 0];
  f32x4 b1 = l4[tid * 2 + 1];

  const float LN2 = 0.69314718055994531f;
  float lg[kPT];
  {
    float ov[kPT] = {o0.x, o0.y, o0.z, o0.w, o1.x, o1.y, o1.z, o1.w};
#pragma unroll
    for (int i = 0; i < kPT; ++i)
      lg[i] = __builtin_amdgcn_logf(ov[i]) * LN2;  // v_log_f32 is log2
  }

  float v[kPT] = {b0.x, b0.y, b0.z, b0.w, b1.x, b1.y, b1.z, b1.w};

  // Thread-local inclusive cumsum.
  float cs[kPT];
  float run = 0.f;
#pragma unroll
  for (int i = 0; i < kPT; ++i) { run += v[i]; cs[i] = run; }

  // Wave32 inclusive scan of per-thread sums (ds_bpermute shuffles).
  float x = run;
#pragma unroll
  for (int d = 1; d < 32; d <<= 1) {
    float y = __shfl_up(x, d, 32);
    if (lane >= d) x += y;
  }
  float texcl = x - run;
  if (lane == 31) s_wsum[wave] = x;
  __syncthreads();

  float wexcl = 0.f, total = 0.f;
#pragma unroll
  for (int i = 0; i < kWaves; ++i) {
    float t = s_wsum[i];
    total += t;
    if (i < wave) wexcl += t;
  }
  float base = wexcl + texcl;

  // F1 simplification: prec=c/k, rec=c/T  =>  F1 = 2c/(k+T), 0 iff c==0.
  const float KEXP = 1.4426950408889634f / TAU_F;  // log2(e)/tau
  float sumE = 0.f, dot = 0.f;
#pragma unroll
  for (int i = 0; i < kPT; ++i) {
    float c = cs[i] + base;
    float k = (float)(tid * kPT + i + 1);
    float r = (c > 0.f) ? (2.f * c) / (k + total) : 0.f;
    float e = __builtin_amdgcn_exp2f(r * KEXP);    // v_exp_f32
    sumE += e;
    dot  += e * lg[i];
  }

  // Deterministic block reduction.
#pragma unroll
  for (int d = 16; d > 0; d >>= 1) {
    sumE += __shfl_down(sumE, d, 32);
    dot  += __shfl_down(dot,  d, 32);
  }
  if (lane == 0) { s_red[wave] = sumE; s_red[kWaves + wave] = dot; }
  __syncthreads();
  if (tid == 0) {
    float S = 0.f, D = 0.f;
#pragma unroll
    for (int i = 0; i < kWaves; ++i) { S += s_red[i]; D += s_red[kWaves + i]; }
    rowpart[row] = D / S;  // per-row  sum_j q_j * log(out_j)
  }
}

// Kernel 2: deterministic fixed-order reduction of 8192 row partials.
__global__ __launch_bounds__(kThreads) void attncut_final_kernel(
    const float* __restrict__ rowpart, float* __restrict__ out)
{
  __shared__ float s[kWaves];
  const int tid  = threadIdx.x;
  const int lane = tid & 31;
  const int wave = tid >> 5;

  float a = 0.f;
  for (int i = tid; i < kB; i += kThreads) a += rowpart[i];
#pragma unroll
  for (int d = 16; d > 0; d >>= 1) a += __shfl_down(a, d, 32);
  if (lane == 0) s[wave] = a;
  __syncthreads();
  if (tid == 0) {
    float t = 0.f;
#pragma unroll
    for (int i = 0; i < kWaves; ++i) t += s[i];
    out[0] = -t / (float)kB;
  }
}

extern "C" void kernel_launch(void* const* d_in, const int* in_sizes, int n_in,
                              void* d_out, int out_size, void* d_ws, size_t ws_size,
                              hipStream_t stream) {
  const float* output = (const float*)d_in[0];  // [B, L, 1] f32
  const float* labels = (const float*)d_in[1];  // [B, L]    f32
  float* rowpart = (float*)d_ws;                // kB floats of scratch
  float* out     = (float*)d_out;               // scalar f32

  attncut_row_kernel<<<kB, kThreads, 0, stream>>>(output, labels, rowpart);
  attncut_final_kernel<<<1, kThreads, 0, stream>>>(rowpart, out);
}